// OSS_50560355008729
// MI455X (gfx1250) — compile-verified
//
#include <hip/hip_runtime.h>
#include <math.h>

#define LSEQ  4096
#define HDIM  64
#define CCH   96
#define PCH   192
#define TWOP  384
#define NST   16
#define DTR   6
#define PROJW 48        // padded 38 -> 48 (3 WMMA N-tiles)
#define NCHUNK 16
#define CLEN  (LSEQ / NCHUNK)   // 256
#define NSEQ  (4 * 2 * PCH)     // 1536 scan sequences

typedef float v2f __attribute__((ext_vector_type(2)));
typedef float v8f __attribute__((ext_vector_type(8)));

__device__ __forceinline__ float silu_f(float v) { return v / (1.0f + __expf(-v)); }

// gather == scatter permutation for each of the 4 scan directions
__device__ __forceinline__ int pix_of(int dir, int l) {
  switch (dir & 3) {
    case 0:  return l;                                   // h-major fwd
    case 1:  return (LSEQ - 1) - l;                      // h-major bwd
    case 2:  return ((l & (HDIM - 1)) << 6) | (l >> 6);  // w-major fwd
    default: { int lf = (LSEQ - 1) - l;                  // w-major bwd
               return ((lf & (HDIM - 1)) << 6) | (lf >> 6); }
  }
}

__global__ __launch_bounds__(256) void k_init_out(float* __restrict__ out,
                                                  const float* __restrict__ x2, int n) {
  int i = blockIdx.x * 256 + threadIdx.x;
  if (i < n) out[i] = x2[i];
}

// xz = gather(x1)[dir] @ W_in[dir]^T ; split into x_raw and silu(z).
// One wave -> 16x64 tile (4 accumulators share each gathered A fragment).
__global__ __launch_bounds__(256) void k_gemm_in(const float* __restrict__ x1,
                                                 const float* __restrict__ Win,
                                                 float* __restrict__ xraw,
                                                 float* __restrict__ zsil) {
  int lane = threadIdx.x & 31;
  int gid  = blockIdx.x * 8 + (threadIdx.x >> 5);   // 0..12287
  int t = lane & 15, h = lane >> 4;
  int ng = gid % 6; int rest = gid / 6;
  int m_t = rest & 255; rest >>= 8;
  int b = rest & 1; int dir = rest >> 1;
  int l0 = m_t << 4, j0 = ng << 6;                  // 64-wide N group

  int pidx = pix_of(dir, l0 + t);
  const float* ap = x1 + ((size_t)b * CCH) * LSEQ + pidx;
  const float* wp = Win + ((size_t)dir * TWOP + (j0 + t)) * CCH;

  v8f acc0 = {0.f,0.f,0.f,0.f,0.f,0.f,0.f,0.f};
  v8f acc1 = acc0, acc2 = acc0, acc3 = acc0;
  for (int k0 = 0; k0 < CCH; k0 += 4) {
    int ka = k0 + (h << 1);
    v2f av, b0, b1, b2, b3;
    av.x = ap[(size_t)ka * LSEQ];
    av.y = ap[(size_t)(ka + 1) * LSEQ];
    b0.x = wp[ka];                b0.y = wp[ka + 1];
    b1.x = wp[16 * CCH + ka];     b1.y = wp[16 * CCH + ka + 1];
    b2.x = wp[32 * CCH + ka];     b2.y = wp[32 * CCH + ka + 1];
    b3.x = wp[48 * CCH + ka];     b3.y = wp[48 * CCH + ka + 1];
    acc0 = __builtin_amdgcn_wmma_f32_16x16x4_f32(false, av, false, b0, (short)0, acc0, false, false);
    acc1 = __builtin_amdgcn_wmma_f32_16x16x4_f32(false, av, false, b1, (short)0, acc1, false, false);
    acc2 = __builtin_amdgcn_wmma_f32_16x16x4_f32(false, av, false, b2, (short)0, acc2, false, false);
    acc3 = __builtin_amdgcn_wmma_f32_16x16x4_f32(false, av, false, b3, (short)0, acc3, false, false);
  }
  size_t rowbase = (size_t)(dir * 2 + b) * LSEQ;
  bool isx = (j0 < PCH);                            // each 64-col group fully x or z
  v8f accs[4] = {acc0, acc1, acc2, acc3};
#pragma unroll
  for (int q = 0; q < 4; ++q) {
    int j = j0 + (q << 4) + t;
#pragma unroll
    for (int r = 0; r < 8; ++r) {
      int l = l0 + r + (h << 3);
      float v = accs[q][r];
      if (isx) xraw[(rowbase + l) * PCH + j] = v;
      else     zsil[(rowbase + l) * PCH + (j - PCH)] = silu_f(v);
    }
  }
}

// depthwise causal conv1d (k=4) + bias + silu
__global__ __launch_bounds__(256) void k_conv(const float* __restrict__ xraw,
                                              const float* __restrict__ convw,
                                              const float* __restrict__ convb,
                                              float* __restrict__ xs) {
  size_t i = (size_t)blockIdx.x * 256 + threadIdx.x;
  if (i >= (size_t)8 * LSEQ * PCH) return;
  int p = (int)(i % PCH); size_t rest = i / PCH;
  int l = (int)(rest % LSEQ); int m = (int)(rest / LSEQ); int dir = m >> 1;
  const float* w  = convw + ((size_t)dir * PCH + p) * 4;
  const float* xr = xraw + ((size_t)m * LSEQ) * PCH + p;
  float s = convb[dir * PCH + p];
#pragma unroll
  for (int k = 0; k < 4; ++k) {
    int ll = l - 3 + k;
    if (ll >= 0) s = fmaf(xr[(size_t)ll * PCH], w[k], s);
  }
  xs[i] = silu_f(s);
}

// proj = xs @ W_x[dir]^T, N padded 38->48. One wave -> 16x48 (A reused 3x).
__global__ __launch_bounds__(256) void k_proj(const float* __restrict__ xs,
                                              const float* __restrict__ Wx,
                                              float* __restrict__ proj) {
  int lane = threadIdx.x & 31;
  int gid  = blockIdx.x * 8 + (threadIdx.x >> 5);   // m_t in [0,2048)
  int t = lane & 15, h = lane >> 4;
  int m0 = gid << 4; int dir = gid >> 9;            // 512 M-tiles per direction
  float msk = (t < 6) ? 1.f : 0.f;                  // tile2 col = 32+t, valid if < 38
  int   jc  = (t < 6) ? (32 + t) : 37;

  const float* ap  = xs + (size_t)(m0 + t) * PCH;
  const float* wp0 = Wx + ((size_t)dir * 38 + t) * PCH;
  const float* wp1 = Wx + ((size_t)dir * 38 + 16 + t) * PCH;
  const float* wp2 = Wx + ((size_t)dir * 38 + jc) * PCH;

  v8f acc0 = {0.f,0.f,0.f,0.f,0.f,0.f,0.f,0.f};
  v8f acc1 = acc0, acc2 = acc0;
  for (int k0 = 0; k0 < PCH; k0 += 4) {
    int ka = k0 + (h << 1);
    v2f av, b0, b1, b2;
    av.x = ap[ka];            av.y = ap[ka + 1];
    b0.x = wp0[ka];           b0.y = wp0[ka + 1];
    b1.x = wp1[ka];           b1.y = wp1[ka + 1];
    b2.x = wp2[ka] * msk;     b2.y = wp2[ka + 1] * msk;
    acc0 = __builtin_amdgcn_wmma_f32_16x16x4_f32(false, av, false, b0, (short)0, acc0, false, false);
    acc1 = __builtin_amdgcn_wmma_f32_16x16x4_f32(false, av, false, b1, (short)0, acc1, false, false);
    acc2 = __builtin_amdgcn_wmma_f32_16x16x4_f32(false, av, false, b2, (short)0, acc2, false, false);
  }
  v8f accs[3] = {acc0, acc1, acc2};
#pragma unroll
  for (int q = 0; q < 3; ++q) {
    int j = (q << 4) + t;
#pragma unroll
    for (int r = 0; r < 8; ++r)
      proj[(size_t)(m0 + r + (h << 3)) * PROJW + j] = accs[q][r];
  }
}

// dt = softplus(proj[:, :6] @ W_dt^T + b_dt), precomputed once (shared by both scan passes)
__global__ __launch_bounds__(256) void k_dt(const float* __restrict__ proj,
                                            const float* __restrict__ Wdt,
                                            const float* __restrict__ bdt,
                                            float* __restrict__ dt) {
  size_t i = (size_t)blockIdx.x * 256 + threadIdx.x;
  if (i >= (size_t)8 * LSEQ * PCH) return;
  int p = (int)(i % PCH); size_t rest = i / PCH;
  int l = (int)(rest % LSEQ); int m = (int)(rest / LSEQ); int dir = m >> 1;
  const float* pr = proj + ((size_t)m * LSEQ + l) * PROJW;
  const float* w  = Wdt + ((size_t)dir * PCH + p) * DTR;
  float s = bdt[dir * PCH + p];
#pragma unroll
  for (int r = 0; r < DTR; ++r) s = fmaf(pr[r], w[r], s);
  dt[i] = (s > 20.f) ? s : log1pf(__expf(s));
}

// Pass 1: per-chunk local scan with h_in = 0; emit carry (h_end, prod dA) per (seq, chunk, n)
__global__ __launch_bounds__(256) void k_scan_local(const float* __restrict__ xs,
                                                    const float* __restrict__ dt,
                                                    const float* __restrict__ proj,
                                                    const float* __restrict__ Alog,
                                                    float* __restrict__ hc,
                                                    float* __restrict__ ac) {
  int tid = blockIdx.x * 256 + threadIdx.x;
  int n = tid & 15;
  int g = tid >> 4;                       // (seq, chunk): 0..24575
  int ch = g % NCHUNK; int s = g / NCHUNK;
  int p = s % PCH; int r2 = s / PCH; int b = r2 & 1; int dir = r2 >> 1;

  float Ac = -__expf(Alog[((size_t)(dir * PCH) + p) * NST + n]);
  size_t rowbase = (size_t)(dir * 2 + b) * LSEQ + (size_t)ch * CLEN;
  const float* xp  = xs + rowbase * PCH + p;
  const float* dtp = dt + rowbase * PCH + p;
  const float* pr  = proj + rowbase * PROJW;

  float hst = 0.f, apr = 1.f;
  for (int l = 0; l < CLEN; ++l) {
    __builtin_prefetch(pr + 4 * PROJW, 0, 1);
    float dtl = dtp[(size_t)l * PCH];
    float xv  = xp[(size_t)l * PCH];
    float Bv  = pr[DTR + n];
    float dA  = __expf(dtl * Ac);
    hst = fmaf(dA, hst, dtl * xv * Bv);
    apr *= dA;
    pr += PROJW;
  }
  hc[(size_t)g * NST + n] = hst;
  ac[(size_t)g * NST + n] = apr;
}

// Pass 2: sequential prefix over the 16 chunk carries per (seq, n) -> h_in per chunk
__global__ __launch_bounds__(256) void k_scan_prefix(const float* __restrict__ hc,
                                                     const float* __restrict__ ac,
                                                     float* __restrict__ hin) {
  int tid = blockIdx.x * 256 + threadIdx.x;   // 0..24575 = (seq, n)
  int n = tid & 15; int s = tid >> 4;
  size_t base = (size_t)s * NCHUNK * NST + n;
  float carry = 0.f;
#pragma unroll
  for (int c = 0; c < NCHUNK; ++c) {
    size_t idx = base + (size_t)c * NST;
    hin[idx] = carry;
    carry = fmaf(ac[idx], carry, hc[idx]);
  }
}

// Pass 3: re-run each chunk seeded with exact prefix h_in; emit y = (<h,C> + x*D) * silu(z)
__global__ __launch_bounds__(256) void k_scan_emit(const float* __restrict__ xs,
                                                   const float* __restrict__ zs,
                                                   const float* __restrict__ dt,
                                                   const float* __restrict__ proj,
                                                   const float* __restrict__ Alog,
                                                   const float* __restrict__ Dp,
                                                   const float* __restrict__ hin,
                                                   float* __restrict__ y) {
  int tid = blockIdx.x * 256 + threadIdx.x;
  int n = tid & 15;
  int g = tid >> 4;                       // (seq, chunk)
  int ch = g % NCHUNK; int s = g / NCHUNK;
  int p = s % PCH; int r2 = s / PCH; int b = r2 & 1; int dir = r2 >> 1;

  float Ac = -__expf(Alog[((size_t)(dir * PCH) + p) * NST + n]);
  float Dv = Dp[dir * PCH + p];
  size_t rowbase = (size_t)(dir * 2 + b) * LSEQ + (size_t)ch * CLEN;
  const float* xp  = xs + rowbase * PCH + p;
  const float* zp  = zs + rowbase * PCH + p;
  const float* dtp = dt + rowbase * PCH + p;
  const float* pr  = proj + rowbase * PROJW;
  float*       yp  = y + rowbase * PCH + p;

  float hst = hin[(size_t)g * NST + n];
  for (int l = 0; l < CLEN; ++l) {
    __builtin_prefetch(pr + 4 * PROJW, 0, 1);
    float dtl = dtp[(size_t)l * PCH];
    float xv  = xp[(size_t)l * PCH];
    float Bv  = pr[DTR + n];
    float Cv  = pr[DTR + NST + n];
    float dA  = __expf(dtl * Ac);
    hst = fmaf(dA, hst, dtl * xv * Bv);
    float acc = hst * Cv;
    acc += __shfl_xor(acc, 8, 16);
    acc += __shfl_xor(acc, 4, 16);
    acc += __shfl_xor(acc, 2, 16);
    acc += __shfl_xor(acc, 1, 16);
    if (n == 0) yp[(size_t)l * PCH] = (acc + xv * Dv) * zp[(size_t)l * PCH];
    pr += PROJW;
  }
}

// out[b,c,pix(dir,l)] += (y @ W_out[dir]^T). One wave -> 16x96 (A reused 6x).
// One launch per dir => race-free, deterministic.
__global__ __launch_bounds__(256) void k_gemm_out(const float* __restrict__ y,
                                                  const float* __restrict__ Wout,
                                                  float* __restrict__ out, int dir) {
  int lane = threadIdx.x & 31;
  int gid  = blockIdx.x * 8 + (threadIdx.x >> 5);   // m_t in [0,512)
  int t = lane & 15, h = lane >> 4;
  int m0 = gid << 4;
  int b  = m0 >> 12; int l0 = m0 & (LSEQ - 1);

  const float* ap = y + ((size_t)(dir * 2 + b) * LSEQ + (l0 + t)) * PCH;
  const float* wp = Wout + ((size_t)dir * CCH + t) * PCH;

  v8f acc0 = {0.f,0.f,0.f,0.f,0.f,0.f,0.f,0.f};
  v8f acc1 = acc0, acc2 = acc0, acc3 = acc0, acc4 = acc0, acc5 = acc0;
  for (int k0 = 0; k0 < PCH; k0 += 4) {
    int ka = k0 + (h << 1);
    v2f av, b0, b1, b2, b3, b4, b5;
    av.x = ap[ka];                 av.y = ap[ka + 1];
    b0.x = wp[ka];                 b0.y = wp[ka + 1];
    b1.x = wp[16 * PCH + ka];      b1.y = wp[16 * PCH + ka + 1];
    b2.x = wp[32 * PCH + ka];      b2.y = wp[32 * PCH + ka + 1];
    b3.x = wp[48 * PCH + ka];      b3.y = wp[48 * PCH + ka + 1];
    b4.x = wp[64 * PCH + ka];      b4.y = wp[64 * PCH + ka + 1];
    b5.x = wp[80 * PCH + ka];      b5.y = wp[80 * PCH + ka + 1];
    acc0 = __builtin_amdgcn_wmma_f32_16x16x4_f32(false, av, false, b0, (short)0, acc0, false, false);
    acc1 = __builtin_amdgcn_wmma_f32_16x16x4_f32(false, av, false, b1, (short)0, acc1, false, false);
    acc2 = __builtin_amdgcn_wmma_f32_16x16x4_f32(false, av, false, b2, (short)0, acc2, false, false);
    acc3 = __builtin_amdgcn_wmma_f32_16x16x4_f32(false, av, false, b3, (short)0, acc3, false, false);
    acc4 = __builtin_amdgcn_wmma_f32_16x16x4_f32(false, av, false, b4, (short)0, acc4, false, false);
    acc5 = __builtin_amdgcn_wmma_f32_16x16x4_f32(false, av, false, b5, (short)0, acc5, false, false);
  }
  v8f accs[6] = {acc0, acc1, acc2, acc3, acc4, acc5};
#pragma unroll
  for (int q = 0; q < 6; ++q) {
    int j = (q << 4) + t;
#pragma unroll
    for (int r = 0; r < 8; ++r) {
      int l = l0 + r + (h << 3);
      int q2 = pix_of(dir, l);
      out[((size_t)b * CCH + j) * LSEQ + q2] += accs[q][r];
    }
  }
}

extern "C" void kernel_launch(void* const* d_in, const int* in_sizes, int n_in,
                              void* d_out, int out_size, void* d_ws, size_t ws_size,
                              hipStream_t stream) {
  (void)in_sizes; (void)n_in; (void)out_size; (void)ws_size;
  const float* x1    = (const float*)d_in[0];
  const float* x2    = (const float*)d_in[1];
  const float* Win   = (const float*)d_in[2];
  const float* convw = (const float*)d_in[3];
  const float* convb = (const float*)d_in[4];
  const float* Wx    = (const float*)d_in[5];
  const float* Wdt   = (const float*)d_in[6];
  const float* bdt   = (const float*)d_in[7];
  const float* Alog  = (const float*)d_in[8];
  const float* Dpv   = (const float*)d_in[9];
  const float* Wout  = (const float*)d_in[10];
  float* out = (float*)d_out;
  float* ws  = (float*)d_ws;

  const size_t nBLP   = (size_t)4 * 2 * LSEQ * PCH;          // 6,291,456 floats
  const size_t nPROJ  = (size_t)4 * 2 * LSEQ * PROJW;        // 1,572,864
  const size_t nCARRY = (size_t)NSEQ * NCHUNK * NST;         //   393,216
  float* xraw  = ws;
  float* zsil  = xraw + nBLP;
  float* xsb   = zsil + nBLP;
  float* projb = xsb  + nBLP;
  float* dtb   = projb + nPROJ;
  float* yb    = dtb  + nBLP;
  float* hc    = yb   + nBLP;
  float* ac    = hc   + nCARRY;
  float* hinb  = ac   + nCARRY;

  const int nOut = 2 * CCH * LSEQ;                           // 786,432
  k_init_out<<<nOut / 256, 256, 0, stream>>>(out, x2, nOut);
  k_gemm_in<<<1536, 256, 0, stream>>>(x1, Win, xraw, zsil);        // 12288 wave-tiles (16x64)
  k_conv<<<24576, 256, 0, stream>>>(xraw, convw, convb, xsb);
  k_proj<<<256, 256, 0, stream>>>(xsb, Wx, projb);                 // 2048 wave-tiles (16x48)
  k_dt<<<24576, 256, 0, stream>>>(projb, Wdt, bdt, dtb);
  k_scan_local<<<1536, 256, 0, stream>>>(xsb, dtb, projb, Alog, hc, ac);
  k_scan_prefix<<<96, 256, 0, stream>>>(hc, ac, hinb);
  k_scan_emit<<<1536, 256, 0, stream>>>(xsb, zsil, dtb, projb, Alog, Dpv, hinb, yb);
  for (int dir = 0; dir < 4; ++dir)
    k_gemm_out<<<64, 256, 0, stream>>>(yb, Wout, out, dir);        // 512 wave-panels (16x96)
}